// RGCN_7395933684254
// MI455X (gfx1250) — compile-verified
//
#include <hip/hip_runtime.h>
#include <hip/hip_bf16.h>

typedef __attribute__((ext_vector_type(2))) float v2f;
typedef __attribute__((ext_vector_type(8))) float v8f;

#define TM 64
#define TN 64
#define KC 16
#define ASTR 20   // padded LDS stride for A tile (conflict-free, 16B aligned)
#define BSTR 80   // padded LDS stride for B tile (conflict-free, 16B aligned)

// C[M,N] = act( A[M,K] @ B[K,N] + bias + addm ), act = LeakyReLU(0.3) if do_lrelu
__launch_bounds__(128)
__global__ void gemm_wmma_f32(const float* __restrict__ A, const float* __restrict__ B,
                              const float* __restrict__ bias, const float* __restrict__ addm,
                              float* __restrict__ C, int M, int N, int K, int do_lrelu)
{
    __shared__ __align__(16) float As[TM * ASTR];
    __shared__ __align__(16) float Bs[KC * BSTR];

    const int t    = threadIdx.x;
    const int wave = t >> 5;
    const int lane = t & 31;
    const int m0   = blockIdx.y * TM;
    const int n0   = blockIdx.x * TN;
    const int half = lane >> 4;     // 0: lanes 0-15, 1: lanes 16-31
    const int lrow = lane & 15;

    v8f acc[4];
    for (int i = 0; i < 4; ++i)
        for (int j = 0; j < 8; ++j) acc[i][j] = 0.0f;

    const int kTiles = (K + KC - 1) / KC;
    for (int kt = 0; kt < kTiles; ++kt) {
        const int k0 = kt * KC;
        // stage A tile 64x16 (256 float4 slots, 2 per thread)
        #pragma unroll
        for (int j = 0; j < 2; ++j) {
            int s   = t * 2 + j;
            int row = s >> 2;
            int c4  = (s & 3) * 4;
            int gm  = m0 + row;
            int gk  = k0 + c4;
            float4 v = make_float4(0.f, 0.f, 0.f, 0.f);
            if (gm < M && (gk + 3) < K)           // all K are multiples of 4
                v = *(const float4*)(A + (size_t)gm * K + gk);
            *(float4*)(&As[row * ASTR + c4]) = v;
        }
        // stage B tile 16x64 (256 float4 slots, 2 per thread)
        #pragma unroll
        for (int j = 0; j < 2; ++j) {
            int s   = t * 2 + j;
            int row = s >> 4;
            int c4  = (s & 15) * 4;
            int gk  = k0 + row;
            int gn  = n0 + c4;
            float4 v = make_float4(0.f, 0.f, 0.f, 0.f);
            if (gk < K && (gn + 3) < N)           // all N are multiples of 16
                v = *(const float4*)(B + (size_t)gk * N + gn);
            *(float4*)(&Bs[row * BSTR + c4]) = v;
        }
        __syncthreads();

        const int am = (wave << 4) + lrow;        // this wave's A row
        #pragma unroll
        for (int kk = 0; kk < 4; ++kk) {
            const int ak = kk * 4 + half * 2;     // A 16x4 layout: lanes16-31 hold K2/K3
            v2f afrag;
            afrag.x = As[am * ASTR + ak];
            afrag.y = As[am * ASTR + ak + 1];
            #pragma unroll
            for (int nt = 0; nt < 4; ++nt) {
                const int bn = (nt << 4) + lrow;  // B 4x16: N striped across lanes
                v2f bfrag;
                bfrag.x = Bs[(kk * 4 + half * 2)     * BSTR + bn];
                bfrag.y = Bs[(kk * 4 + half * 2 + 1) * BSTR + bn];
                acc[nt] = __builtin_amdgcn_wmma_f32_16x16x4_f32(
                    false, afrag, false, bfrag, (short)0, acc[nt], false, false);
            }
        }
        __syncthreads();
    }

    // epilogue: C/D layout -> VGPR v holds rows (half*8 + v), col = n0+16*nt+lrow
    #pragma unroll
    for (int nt = 0; nt < 4; ++nt) {
        int gn = n0 + (nt << 4) + lrow;
        if (gn >= N) continue;
        float bv = bias ? bias[gn] : 0.0f;
        #pragma unroll
        for (int v = 0; v < 8; ++v) {
            int gm = m0 + (wave << 4) + half * 8 + v;
            if (gm >= M) continue;
            float val = acc[nt][v] + bv;
            if (addm) val += addm[(size_t)gm * N + gn];
            if (do_lrelu) val = (val > 0.0f) ? val : 0.3f * val;
            C[(size_t)gm * N + gn] = val;
        }
    }
}

// agg[dst[e]] += proj[src[e]] for edges of relation `rel`; one wave per edge,
// lane i handles dwords i, i+32, ... (coalesced loads + coalesced f32 atomics).
__global__ void scatter_rel(const float* __restrict__ proj, const int* __restrict__ src,
                            const int* __restrict__ dst, const int* __restrict__ et,
                            int rel, int nE, float* __restrict__ agg)
{
    int wid  = (int)((blockIdx.x * blockDim.x + threadIdx.x) >> 5);
    int lane = threadIdx.x & 31;
    if (wid >= nE) return;
    if (et[wid] != rel) return;
    const float* p = proj + (size_t)src[wid] * 256;
    float*       a = agg  + (size_t)dst[wid] * 256;
    #pragma unroll
    for (int j = 0; j < 8; ++j)
        atomicAdd(a + lane + 32 * j, p[lane + 32 * j]);
}

__global__ void loss_kernel(const float* __restrict__ cf, const float* __restrict__ feat,
                            float* __restrict__ out_loss, int n)
{
    __shared__ float red[256];
    int idx = blockIdx.x * blockDim.x + threadIdx.x;
    float s = 0.0f;
    for (int i = idx; i < n; i += gridDim.x * blockDim.x) {
        float d = cf[i] - feat[i];
        s += d * d;
    }
    red[threadIdx.x] = s;
    __syncthreads();
    for (int off = 128; off > 0; off >>= 1) {
        if ((int)threadIdx.x < off) red[threadIdx.x] += red[threadIdx.x + off];
        __syncthreads();
    }
    if (threadIdx.x == 0) atomicAdd(out_loss, red[0] / (float)n);
}

// res[i] = mean over the 3 node-type blocks of cf (blocks of 20000*16 floats)
__global__ void mean_kernel(const float* __restrict__ cf, float* __restrict__ out, int per)
{
    int i = blockIdx.x * blockDim.x + threadIdx.x;
    if (i < per)
        out[i] = (cf[i] + cf[i + per] + cf[i + 2 * per]) * (1.0f / 3.0f);
}

extern "C" void kernel_launch(void* const* d_in, const int* in_sizes, int n_in,
                              void* d_out, int out_size, void* d_ws, size_t ws_size,
                              hipStream_t stream)
{
    const float* feat0 = (const float*)d_in[0];   // [20000,100]
    const float* feat1 = (const float*)d_in[1];   // [20000,300]
    const float* feat2 = (const float*)d_in[2];   // [20000,768]
    const int*   src   = (const int*)d_in[3];
    const int*   dst   = (const int*)d_in[4];
    const int*   et    = (const int*)d_in[5];
    const float* e0w1 = (const float*)d_in[6];  const float* e0b1 = (const float*)d_in[7];
    const float* e0w2 = (const float*)d_in[8];  const float* e0b2 = (const float*)d_in[9];
    const float* e1w1 = (const float*)d_in[10]; const float* e1b1 = (const float*)d_in[11];
    const float* e1w2 = (const float*)d_in[12]; const float* e1b2 = (const float*)d_in[13];
    const float* e2w1 = (const float*)d_in[14]; const float* e2b1 = (const float*)d_in[15];
    const float* e2w2 = (const float*)d_in[16]; const float* e2b2 = (const float*)d_in[17];
    const float* e2w3 = (const float*)d_in[18]; const float* e2b3 = (const float*)d_in[19];
    const float* gW   = (const float*)d_in[20]; // [2,5,256,256]
    const float* gL   = (const float*)d_in[21]; // [2,256,256]
    const float* gB   = (const float*)d_in[22]; // [2,256]
    const float* dw1  = (const float*)d_in[23]; const float* db1 = (const float*)d_in[24];
    const float* dw2  = (const float*)d_in[25]; const float* db2 = (const float*)d_in[26];
    const float* cew1 = (const float*)d_in[27]; const float* ceb1 = (const float*)d_in[28];
    const float* cew2 = (const float*)d_in[29]; const float* ceb2 = (const float*)d_in[30];
    const float* cdw1 = (const float*)d_in[31]; const float* cdb1 = (const float*)d_in[32];
    const float* cdw2 = (const float*)d_in[33]; const float* cdb2 = (const float*)d_in[34];

    const int E = in_sizes[3];
    const int NPER = 20000, NNODES = 60000, H = 256;
    const size_t NH = (size_t)NNODES * H;          // 15,360,000 floats

    float* ws   = (float*)d_ws;
    float* h    = ws;                               // [60000,256]
    float* hnew = ws + NH;                          // [60000,256]
    float* agg  = ws + 2 * NH;                      // [60000,256]
    float* tmp  = ws + 3 * NH;                      // [<=60000,256] scratch
    float* big  = ws + 4 * NH;                      // [60000,1024] scratch
    float* feat = big + (size_t)NNODES * 1024;      // [60000,16]
    float* cf   = feat + (size_t)NNODES * 16;       // [60000,16]

    auto gemm = [&](const float* A, const float* B, const float* bias, const float* addm,
                    float* C, int M, int N, int K, int lrelu) {
        dim3 grid((N + TN - 1) / TN, (M + TM - 1) / TM);
        gemm_wmma_f32<<<grid, 128, 0, stream>>>(A, B, bias, addm, C, M, N, K, lrelu);
    };

    // ---- per-type encoders -> h [60000,256] ----
    gemm(feat0, e0w1, e0b1, nullptr, tmp, NPER, 256, 100, 1);
    gemm(tmp,   e0w2, e0b2, nullptr, h,                      NPER, 256, 256, 0);
    gemm(feat1, e1w1, e1b1, nullptr, tmp, NPER, 256, 300, 1);
    gemm(tmp,   e1w2, e1b2, nullptr, h + (size_t)NPER * H,   NPER, 256, 256, 0);
    gemm(feat2, e2w1, e2b1, nullptr, big, NPER, 512, 768, 1);
    gemm(big,   e2w2, e2b2, nullptr, tmp, NPER, 256, 512, 1);
    gemm(tmp,   e2w3, e2b3, nullptr, h + (size_t)2 * NPER * H, NPER, 256, 256, 0);

    // ---- RelGraphConv layers ----
    float* hc = h; float* hn = hnew;
    for (int l = 0; l < 2; ++l) {
        hipMemsetAsync(agg, 0, NH * sizeof(float), stream);
        for (int r = 0; r < 5; ++r) {
            gemm(hc, gW + (size_t)(l * 5 + r) * H * H, nullptr, nullptr, tmp, NNODES, 256, 256, 0);
            dim3 g((E + 7) / 8);   // 8 waves (edges) per 256-thread block
            scatter_rel<<<g, 256, 0, stream>>>(tmp, src, dst, et, r, E, agg);
        }
        gemm(hc, gL + (size_t)l * H * H, gB + (size_t)l * H, agg, hn, NNODES, 256, 256, 0);
        float* sw = hc; hc = hn; hn = sw;
    }

    // ---- decoder: feat = lrelu(h@dw1+db1)@dw2+db2 ----
    gemm(hc,  dw1, db1, nullptr, tmp,  NNODES, 256, 256, 1);
    gemm(tmp, dw2, db2, nullptr, feat, NNODES, 16,  256, 0);
    // ---- constr encoder/decoder ----
    gemm(feat, cew1, ceb1, nullptr, big, NNODES, 1024, 16,   1);
    gemm(big,  cew2, ceb2, nullptr, tmp, NNODES, 256,  1024, 0);
    gemm(tmp,  cdw1, cdb1, nullptr, big, NNODES, 256,  256,  1); // big reused as [60000,256]
    gemm(big,  cdw2, cdb2, nullptr, cf,  NNODES, 16,   256,  0);

    // ---- outputs: res [20000,16] then con_loss scalar ----
    float* out = (float*)d_out;
    hipMemsetAsync(out, 0, (size_t)(NPER * 16 + 1) * sizeof(float), stream);
    loss_kernel<<<512, 256, 0, stream>>>(cf, feat, out + NPER * 16, NNODES * 16);
    mean_kernel<<<(NPER * 16 + 255) / 256, 256, 0, stream>>>(cf, out, NPER * 16);
}